// GCN_more_layer_VI_58248346468473
// MI455X (gfx1250) — compile-verified
//
#include <hip/hip_runtime.h>
#include <cstdint>
#include <cstddef>

// ---------------------------------------------------------------------------
// GCN 3-layer forward for MI455X (gfx1250).
//  - fp32 WMMA (v_wmma_f32_16x16x4_f32) for the dense per-layer GEMMs
//  - L2-resident gather + global_atomic_add_f32 scatter for edge aggregation
// ---------------------------------------------------------------------------

typedef float v2f __attribute__((ext_vector_type(2)));
typedef float v8f __attribute__((ext_vector_type(8)));

#define FEAT 128
#define TB   256

__device__ __forceinline__ void atomAddF(float* p, float v) {
    // HW f32 atomic add (L2-resident accumulation); no CAS loop.
    unsafeAtomicAdd(p, v);
}

// ----------------------------- degree / norm -------------------------------

__global__ void k_deg_init(float* __restrict__ deg, int n) {
    int i = blockIdx.x * blockDim.x + threadIdx.x;
    if (i < n) deg[i] = 1.0f;                     // self-loop contribution
}

__global__ void k_deg_edges(const long long* __restrict__ dst,
                            float* __restrict__ deg, int e) {
    int i = blockIdx.x * blockDim.x + threadIdx.x;
    int stride = gridDim.x * blockDim.x;
    for (; i < e; i += stride) atomAddF(&deg[(int)dst[i]], 1.0f);
}

__global__ void k_dinv(float* __restrict__ d, int n) {
    int i = blockIdx.x * blockDim.x + threadIdx.x;
    if (i < n) d[i] = rsqrtf(d[i]);               // deg >= 1 always (self-loop)
}

// ------------------------------- WMMA GEMM ---------------------------------
// Out[N x 128] = H[N x 128] @ W[128 x 128]
// Block: 256 threads = 8 waves. Block owns 16 rows; wave w owns cols [16w,16w+16).
// A tile staged in LDS with pitch 132 floats -> conflict-free ds_load_b64.
// WMMA f32 16x16x4 layout: lane&15 = M (A) / N (B); reg r holds K = 2*(lane>>4)+r.

#define APITCH 132

__global__ __launch_bounds__(TB) void k_gemm128_wmma(
        const float* __restrict__ H, const float* __restrict__ Wm,
        float* __restrict__ Out, int n) {
    __shared__ __align__(16) float At[16 * APITCH];

    const int t       = threadIdx.x;
    const int rowbase = blockIdx.x * 16;

    // Cooperative, coalesced stage of the 16x128 A tile (float4 per thread x2)
    for (int i = t; i < (16 * FEAT) / 4; i += TB) {
        int off = i * 4;
        int r = off >> 7;          // /128
        int c = off & 127;
        int gr = rowbase + r; if (gr > n - 1) gr = n - 1;   // clamp, keep EXEC full
        *(float4*)(At + r * APITCH + c) =
            *(const float4*)(H + (size_t)gr * FEAT + c);
    }
    __syncthreads();

    const int lane = t & 31;
    const int half = lane >> 4;                  // 0 | 1
    const int lm   = lane & 15;                  // M (A) / N (B/C/D)
    const int col  = ((t >> 5) << 4) + lm;       // wave's output column

    v8f acc = {0.f, 0.f, 0.f, 0.f, 0.f, 0.f, 0.f, 0.f};

    #pragma unroll
    for (int k0 = 0; k0 < FEAT; k0 += 4) {
        const int kk = k0 + 2 * half;            // reg r -> K = kk + r
        v2f a = *(const v2f*)(At + lm * APITCH + kk);   // 8B LDS read, no conflicts
        v2f b;
        b.x = Wm[(size_t)kk * FEAT + col];
        b.y = Wm[(size_t)(kk + 1) * FEAT + col];
        acc = __builtin_amdgcn_wmma_f32_16x16x4_f32(
                  false, a, false, b, (short)0, acc, false, false);
    }

    // D layout: reg v -> row M = v + 8*half, lane&15 -> col
    #pragma unroll
    for (int v = 0; v < 8; ++v) {
        int row = rowbase + v + 8 * half;
        if (row < n) Out[(size_t)row * FEAT + col] = acc[v];
    }
}

// ------------------------- layer-3 GEMV (128 -> 1) -------------------------

__global__ __launch_bounds__(TB) void k_gemv128(
        const float* __restrict__ H, const float* __restrict__ w,
        float* __restrict__ out, int n) {
    const int lane = threadIdx.x & 31;
    const int row  = blockIdx.x * (blockDim.x >> 5) + (threadIdx.x >> 5);
    if (row >= n) return;
    float4 hv = *(const float4*)(H + (size_t)row * FEAT + lane * 4);
    float4 wv = *(const float4*)(w + lane * 4);
    float s = hv.x * wv.x + hv.y * wv.y + hv.z * wv.z + hv.w * wv.w;
    #pragma unroll
    for (int m = 16; m >= 1; m >>= 1) s += __shfl_xor(s, m, 32);
    if (lane == 0) out[row] = s;
}

// ------------------- self-loop init (+bias) = output init ------------------

__global__ void k_init_self128(const float* __restrict__ hw,
                               const float* __restrict__ dinv,
                               const float* __restrict__ bias,
                               float* __restrict__ out, int n) {
    int i = blockIdx.x * blockDim.x + threadIdx.x;   // float4 index
    if (i >= n * 32) return;
    int node = i >> 5;
    int c4   = (i & 31) * 4;
    float di = dinv[node];
    float sc = di * di;
    float4 v = *(const float4*)(hw + (size_t)i * 4);
    float4 b = *(const float4*)(bias + c4);
    float4 o;
    o.x = v.x * sc + b.x;  o.y = v.y * sc + b.y;
    o.z = v.z * sc + b.z;  o.w = v.w * sc + b.w;
    *(float4*)(out + (size_t)i * 4) = o;
}

__global__ void k_init_self1(const float* __restrict__ hw,
                             const float* __restrict__ dinv,
                             const float* __restrict__ bias,
                             float* __restrict__ out, int n) {
    int i = blockIdx.x * blockDim.x + threadIdx.x;
    if (i < n) { float di = dinv[i]; out[i] = hw[i] * di * di + bias[0]; }
}

// -------------------------- edge scatter-aggregate -------------------------
// One wave32 per edge: float4 gather per lane (coalesced 512B from L2),
// 4 hardware f32 atomic adds per lane into the destination row.

__global__ __launch_bounds__(TB) void k_edge_agg128(
        const float* __restrict__ hw,
        const long long* __restrict__ src, const long long* __restrict__ dst,
        const float* __restrict__ dinv, float* __restrict__ out, int e) {
    const int lane = threadIdx.x & 31;
    int       w    = (blockIdx.x * blockDim.x + threadIdx.x) >> 5;
    const int nw   = (gridDim.x * blockDim.x) >> 5;
    for (; w < e; w += nw) {
        const int   s   = (int)src[w];
        const int   d   = (int)dst[w];
        const float nrm = dinv[s] * dinv[d];
        float4 v = *(const float4*)(hw + (size_t)s * FEAT + lane * 4);
        float* o = out + (size_t)d * FEAT + lane * 4;
        atomAddF(o + 0, v.x * nrm);
        atomAddF(o + 1, v.y * nrm);
        atomAddF(o + 2, v.z * nrm);
        atomAddF(o + 3, v.w * nrm);
    }
}

__global__ void k_edge_agg1(const float* __restrict__ hw,
                            const long long* __restrict__ src,
                            const long long* __restrict__ dst,
                            const float* __restrict__ dinv,
                            float* __restrict__ out, int e) {
    int i = blockIdx.x * blockDim.x + threadIdx.x;
    int stride = gridDim.x * blockDim.x;
    for (; i < e; i += stride) {
        int s = (int)src[i], d = (int)dst[i];
        atomAddF(&out[d], hw[s] * dinv[s] * dinv[d]);
    }
}

// ------------------------------ activations --------------------------------

__global__ void k_relu4(float* __restrict__ x, int n4) {
    int i = blockIdx.x * blockDim.x + threadIdx.x;
    if (i >= n4) return;
    float4 v = *(float4*)(x + (size_t)i * 4);
    v.x = fmaxf(v.x, 0.f); v.y = fmaxf(v.y, 0.f);
    v.z = fmaxf(v.z, 0.f); v.w = fmaxf(v.w, 0.f);
    *(float4*)(x + (size_t)i * 4) = v;
}

__global__ void k_sigmoid(float* __restrict__ x, int n) {
    int i = blockIdx.x * blockDim.x + threadIdx.x;
    if (i < n) x[i] = 1.0f / (1.0f + expf(-x[i]));
}

// ------------------------------- launcher ----------------------------------

extern "C" void kernel_launch(void* const* d_in, const int* in_sizes, int n_in,
                              void* d_out, int out_size, void* d_ws, size_t ws_size,
                              hipStream_t stream) {
    const float*     x   = (const float*)d_in[0];
    const long long* ei  = (const long long*)d_in[1];
    const float*     W1  = (const float*)d_in[2];
    const float*     b1  = (const float*)d_in[3];
    const float*     W2  = (const float*)d_in[4];
    const float*     b2  = (const float*)d_in[5];
    const float*     W3  = (const float*)d_in[6];
    const float*     b3  = (const float*)d_in[7];

    const int N = in_sizes[0] / FEAT;
    const int E = in_sizes[1] / 2;
    const long long* src = ei;          // edge_index[0]
    const long long* dst = ei + E;      // edge_index[1]

    // Workspace: dinv[N] | hw[N*128] | hb[N*128]   (~103 MB)
    char*  ws   = (char*)d_ws;
    float* dinv = (float*)ws;
    size_t off  = ((size_t)N * 4 + 255) & ~(size_t)255;
    float* hw   = (float*)(ws + off);
    off        += (((size_t)N * FEAT * 4) + 255) & ~(size_t)255;
    float* hb   = (float*)(ws + off);

    const int gN    = (N + TB - 1) / TB;
    const int gNF4  = (N * 32 + TB - 1) / TB;
    const int gGemm = (N + 15) / 16;
    const int gGemv = (N + 7) / 8;
    const int gEdgeT = 2048;   // thread-per-edge grid-stride
    const int gEdgeW = 4096;   // wave-per-edge grid-stride

    // --- normalization ---
    k_deg_init <<<gN,     TB, 0, stream>>>(dinv, N);
    k_deg_edges<<<gEdgeT, TB, 0, stream>>>(dst, dinv, E);
    k_dinv     <<<gN,     TB, 0, stream>>>(dinv, N);

    // --- layer 1: relu(A_hat @ (x @ W1) + b1) ---
    k_gemm128_wmma<<<gGemm,  TB, 0, stream>>>(x, W1, hw, N);
    k_init_self128<<<gNF4,   TB, 0, stream>>>(hw, dinv, b1, hb, N);
    k_edge_agg128 <<<gEdgeW, TB, 0, stream>>>(hw, src, dst, dinv, hb, E);
    k_relu4       <<<gNF4,   TB, 0, stream>>>(hb, N * 32);

    // --- layer 2 ---
    k_gemm128_wmma<<<gGemm,  TB, 0, stream>>>(hb, W2, hw, N);
    k_init_self128<<<gNF4,   TB, 0, stream>>>(hw, dinv, b2, hb, N);
    k_edge_agg128 <<<gEdgeW, TB, 0, stream>>>(hw, src, dst, dinv, hb, E);
    k_relu4       <<<gNF4,   TB, 0, stream>>>(hb, N * 32);

    // --- layer 3: sigmoid(A_hat @ (h2 @ W3) + b3) ---
    float* outp = (float*)d_out;
    k_gemv128   <<<gGemv,  TB, 0, stream>>>(hb, W3, hw, N);
    k_init_self1<<<gN,     TB, 0, stream>>>(hw, dinv, b3, outp, N);
    k_edge_agg1 <<<gEdgeT, TB, 0, stream>>>(hw, src, dst, dinv, outp, E);
    k_sigmoid   <<<gN,     TB, 0, stream>>>(outp, N);
}